// NeighborhoodAttention1D_60198261621536
// MI455X (gfx1250) — compile-verified
//
#include <hip/hip_runtime.h>
#include <hip/hip_bf16.h>

// ---------------- problem constants (from reference) ----------------
#define BATCH   4
#define SEQL    4096
#define CH      512          // C
#define NHEAD   16
#define HDIM    32           // C / H
#define KSZ     13
#define KHALF   6            // KSZ/2
#define RPBW    25           // 2*K-1
#define ROWS    (BATCH * SEQL)   // 16384

typedef __attribute__((ext_vector_type(2))) float v2f;
typedef __attribute__((ext_vector_type(8))) float v8f;
typedef __attribute__((ext_vector_type(4))) unsigned int v4u;
typedef __attribute__((ext_vector_type(8))) int v8i;
typedef __attribute__((ext_vector_type(4))) int v4i;

// =====================================================================
// TDM descriptor issue: 2D tile (rows x cols of fp32) global -> LDS,
// with LDS row padding (reproduces the bank-conflict-free strides).
//   d0_flags = data_size | pad_enable | pad_interval | pad_amount
// =====================================================================
__device__ __forceinline__ void tdm_load_2d(unsigned lds_addr,
                                            const void* gaddr,
                                            unsigned d0_flags,
                                            unsigned tensor_d0, unsigned tensor_d1,
                                            unsigned tile_d0,   unsigned tile_d1,
                                            unsigned long long stride0 /*elements*/)
{
    unsigned long long ga = (unsigned long long)gaddr;

    v4u g0;
    g0[0] = 1u;                                          // count=1 (user D#)
    g0[1] = lds_addr;                                    // LDS byte address
    g0[2] = (unsigned)(ga & 0xFFFFFFFFu);                // global_addr[31:0]
    g0[3] = (unsigned)((ga >> 32) & 0x01FFFFFFu)         // global_addr[56:32]
          | (2u << 30);                                  // type = 2 ("image")

    v8i g1;
    g1[0] = (int)d0_flags;                               // mask=0|dsize|pad cfg
    g1[1] = (int)((tensor_d0 & 0xFFFFu) << 16);          // tensor_dim0[15:0]
    g1[2] = (int)((tensor_d0 >> 16) | ((tensor_d1 & 0xFFFFu) << 16));
    g1[3] = (int)((tensor_d1 >> 16) | (tile_d0 << 16));  // tile_dim0
    g1[4] = (int)(tile_d1 & 0xFFFFu);                    // tile_dim1, tile_dim2=0
    g1[5] = (int)(stride0 & 0xFFFFFFFFu);                // tensor_dim0_stride
    g1[6] = (int)((stride0 >> 32) & 0xFFFFu);            // (dim1_stride unused)
    g1[7] = 0;

    v4i z4 = {0, 0, 0, 0};
#if defined(__clang_major__) && (__clang_major__ >= 23)
    v8i z8 = {0, 0, 0, 0, 0, 0, 0, 0};
    __builtin_amdgcn_tensor_load_to_lds(g0, g1, z4, z4, z8, 0);
#else
    __builtin_amdgcn_tensor_load_to_lds(g0, g1, z4, z4, 0);
#endif
}

// data_size=4B (2<<16) | pad_enable (1<<20) | pad_interval | pad_amount
#define D0_A ((2u << 16) | (1u << 20) | (3u << 22) | (3u << 25))  // 16 DW rows, +4 DW pad -> stride 20
#define D0_B ((2u << 16) | (1u << 20) | (5u << 22) | (7u << 25))  // 64 DW rows, +8 DW pad -> stride 72

// =====================================================================
// FP32 WMMA GEMM:  C[M,N] = A[M,K] * B[K,N] + bias[N]
//   64x64 block tile, 4 wave32s -> 32x32 each (2x2 WMMA 16x16x4).
//   K staged through double-buffered LDS via the Tensor Data Mover.
// =====================================================================
#define TM 64
#define TN 64
#define TK 16
#define AS_STRIDE 20
#define BS_STRIDE 72
#define AS_ELEMS (TM * AS_STRIDE)   // 1280 floats = 5120 B per buffer
#define BS_ELEMS (TK * BS_STRIDE)   // 1152 floats = 4608 B per buffer

__global__ __launch_bounds__(128)
void wmma_gemm_f32(const float* __restrict__ A,
                   const float* __restrict__ Bm,
                   const float* __restrict__ bias,
                   float* __restrict__ C,
                   int M, int N, int K)
{
    __shared__ float As[2 * AS_ELEMS];
    __shared__ float Bs[2 * BS_ELEMS];

    const int tid  = threadIdx.x;
    const int lane = tid & 31;
    const int wave = tid >> 5;              // 0..3
    const int wm   = (wave >> 1) * 32;
    const int wn   = (wave & 1)  * 32;
    const int lrow = lane & 15;
    const int lhi  = lane >> 4;

    const int bm = blockIdx.y * TM;
    const int bn = blockIdx.x * TN;

    // LDS byte addresses (low 32 bits of flat addr == LDS offset)
    const unsigned as_base = (unsigned)(unsigned long long)(&As[0]);
    const unsigned bs_base = (unsigned)(unsigned long long)(&Bs[0]);

    const bool tdm_wave = (wave == 0);      // TDM ignores EXEC; one wave issues
    const int  nslab    = K / TK;

    auto issue_slab = [&](int s) {
        const int k0 = s * TK;
        const int p  = s & 1;
        // A slab: 64 rows x 16 cols from row-major MxK
        tdm_load_2d(as_base + (unsigned)p * (AS_ELEMS * 4),
                    A + (size_t)bm * K + k0,
                    D0_A, (unsigned)K, (unsigned)M, TK, TM,
                    (unsigned long long)K);
        // B slab: 16 rows x 64 cols from row-major KxN
        tdm_load_2d(bs_base + (unsigned)p * (BS_ELEMS * 4),
                    Bm + (size_t)k0 * N + bn,
                    D0_B, (unsigned)N, (unsigned)K, TN, TK,
                    (unsigned long long)N);
    };

    v8f acc[2][2] = {};

    if (tdm_wave) issue_slab(0);

    for (int s = 0; s < nslab; ++s) {
        if (tdm_wave) {
            if (s + 1 < nslab) {
                issue_slab(s + 1);                       // prefetch next slab
                __builtin_amdgcn_s_wait_tensorcnt(2);    // slab s complete
            } else {
                __builtin_amdgcn_s_wait_tensorcnt(0);    // final slab complete
            }
        }
        __syncthreads();   // LDS data visible to all waves

        const float* asb = &As[(s & 1) * AS_ELEMS];
        const float* bsb = &Bs[(s & 1) * BS_ELEMS];

        #pragma unroll
        for (int kk = 0; kk < TK; kk += 4) {
            const int ksub = kk + 2 * lhi;               // K base for lane half
            v2f afrag[2], bfrag[2];
            #pragma unroll
            for (int t = 0; t < 2; ++t) {
                int m = wm + t * 16 + lrow;              // A frag: VGPR j <-> K=ksub+j
                afrag[t].x = asb[m * AS_STRIDE + ksub];
                afrag[t].y = asb[m * AS_STRIDE + ksub + 1];
                int n = wn + t * 16 + lrow;              // B frag: VGPR j <-> K=ksub+j
                bfrag[t].x = bsb[(ksub)     * BS_STRIDE + n];
                bfrag[t].y = bsb[(ksub + 1) * BS_STRIDE + n];
            }
            #pragma unroll
            for (int tm = 0; tm < 2; ++tm)
                #pragma unroll
                for (int tn = 0; tn < 2; ++tn)
                    acc[tm][tn] = __builtin_amdgcn_wmma_f32_16x16x4_f32(
                        false, afrag[tm], false, bfrag[tn],
                        (short)0, acc[tm][tn], false, false);
        }
        __syncthreads();   // all waves done reading buffer (s&1) before reuse
    }

    // ---- store 2x2 fragments + bias; C/D layout: VGPR j -> M = j + 8*lhi ----
    #pragma unroll
    for (int tm = 0; tm < 2; ++tm)
        #pragma unroll
        for (int tn = 0; tn < 2; ++tn) {
            int n = bn + wn + tn * 16 + lrow;
            float bv = bias[n];
            #pragma unroll
            for (int j = 0; j < 8; ++j) {
                int m = bm + wm + tm * 16 + j + 8 * lhi;
                C[(size_t)m * N + n] = acc[tm][tn][j] + bv;
            }
        }
}

// =====================================================================
// Neighborhood attention: one wave32 per QPW (8) queries, lane = head dim.
// =====================================================================
#define QPW 8

__global__ __launch_bounds__(256)
void natten_kernel(const float* __restrict__ qkv,
                   const float* __restrict__ rpb,
                   float* __restrict__ out)
{
    const int lane  = threadIdx.x & 31;
    const int wave  = threadIdx.x >> 5;
    const int gwave = blockIdx.x * 8 + wave;

    const int q0 = gwave * QPW;
    const int bh = q0 / SEQL;
    const int l0 = q0 % SEQL;
    const int b  = bh / NHEAD;
    const int h  = bh % NHEAD;

    const float scale = 0.17677669529663687f;   // 1/sqrt(32)
    const size_t rowstride = 3 * CH;
    const int coff = h * HDIM + lane;

    for (int i = 0; i < QPW; ++i) {
        const int l = l0 + i;
        int n0 = l - KHALF;
        n0 = (n0 < 0) ? 0 : n0;
        n0 = (n0 > SEQL - KSZ) ? (SEQL - KSZ) : n0;

        const size_t qrow = (size_t)(b * SEQL + l) * rowstride;
        const float qd = qkv[qrow + coff] * scale;

        float p[KSZ];
        float mx = -3.4e38f;
        const int rpb_base = h * RPBW + (n0 - l + (KSZ - 1));
        #pragma unroll
        for (int kk = 0; kk < KSZ; ++kk) {
            const size_t krow = (size_t)(b * SEQL + n0 + kk) * rowstride;
            float t = qd * qkv[krow + CH + coff];
            #pragma unroll
            for (int off = 16; off > 0; off >>= 1)
                t += __shfl_xor(t, off, 32);
            t += rpb[rpb_base + kk];
            p[kk] = t;
            mx = fmaxf(mx, t);
        }

        float denom = 0.f;
        #pragma unroll
        for (int kk = 0; kk < KSZ; ++kk) {
            p[kk] = __expf(p[kk] - mx);
            denom += p[kk];
        }
        const float inv = 1.f / denom;

        float o = 0.f;
        #pragma unroll
        for (int kk = 0; kk < KSZ; ++kk) {
            const size_t vrow = (size_t)(b * SEQL + n0 + kk) * rowstride;
            o += p[kk] * qkv[vrow + 2 * CH + coff];
        }
        out[(size_t)(b * SEQL + l) * CH + coff] = o * inv;
    }
}

// =====================================================================
extern "C" void kernel_launch(void* const* d_in, const int* in_sizes, int n_in,
                              void* d_out, int out_size, void* d_ws, size_t ws_size,
                              hipStream_t stream)
{
    const float* x      = (const float*)d_in[0];
    const float* qkv_w  = (const float*)d_in[1];
    const float* qkv_b  = (const float*)d_in[2];
    const float* rpb    = (const float*)d_in[3];
    const float* proj_w = (const float*)d_in[4];
    const float* proj_b = (const float*)d_in[5];
    float* out = (float*)d_out;

    float* qkv_ws  = (float*)d_ws;                         // 16384 x 1536
    float* attn_ws = qkv_ws + (size_t)ROWS * (3 * CH);     // 16384 x 512

    wmma_gemm_f32<<<dim3((3 * CH) / TN, ROWS / TM), dim3(128), 0, stream>>>(
        x, qkv_w, qkv_b, qkv_ws, ROWS, 3 * CH, CH);

    natten_kernel<<<dim3((BATCH * NHEAD * SEQL) / (8 * QPW)), dim3(256), 0, stream>>>(
        qkv_ws, rpb, attn_ws);

    wmma_gemm_f32<<<dim3(CH / TN, ROWS / TM), dim3(128), 0, stream>>>(
        attn_ws, proj_w, proj_b, out, ROWS, CH, CH);
}